// CausalGCN_59150289601188
// MI455X (gfx1250) — compile-verified
//
#include <hip/hip_runtime.h>
#include <math.h>

typedef __attribute__((ext_vector_type(2))) float v2f;
typedef __attribute__((ext_vector_type(8))) float v8f;

#define HDIM 128
#define GRP  128
#define EPSV 1e-5f
// LDS layout for W: j-row (K-pair) stride, padded so lanes 0-15 (pair j) and
// 16-31 (pair j+1) hit disjoint bank halves: 288 mod 64 == 32.
#define WP_STRIDE 288

// ---------------- init / degree / normalization --------------------------
__global__ void k_init(float* deg, float* stats, float* cnt, float* xc, float* xo,
                       int n, int gh) {
    int i = blockIdx.x * blockDim.x + threadIdx.x;
    if (i < n)   deg[i] = 1.0f;          // self loop contributes 1 to degree
    if (i < 256) stats[i] = 0.0f;
    if (i < GRP) cnt[i] = 0.0f;
    if (i < gh)  { xc[i] = 0.0f; xo[i] = 0.0f; }
}

__global__ void k_degree(const int* __restrict__ row, float* __restrict__ deg, int E) {
    int e = blockIdx.x * blockDim.x + threadIdx.x;
    if (e < E) atomicAdd(&deg[row[e]], 1.0f);
}

__global__ void k_rsqrt(float* d, int n) {
    int i = blockIdx.x * blockDim.x + threadIdx.x;
    if (i < n) d[i] = rsqrtf(fmaxf(d[i], 1.0f));
}

// ---------------- input BatchNorm ----------------------------------------
__global__ void k_bnstats(const float* __restrict__ x, float* __restrict__ stats, int n) {
    int f = threadIdx.x;                                  // 128 threads
    float s = 0.0f, sq = 0.0f;
    for (int i = blockIdx.x; i < n; i += gridDim.x) {
        float v = x[(size_t)i * HDIM + f];
        s += v; sq += v * v;
    }
    atomicAdd(&stats[f], s);
    atomicAdd(&stats[HDIM + f], sq);
}

__global__ void k_bnapply(const float* __restrict__ x, const float* __restrict__ stats,
                          const float* __restrict__ g, const float* __restrict__ b,
                          float* __restrict__ h, int n) {
    int i = blockIdx.x * blockDim.x + threadIdx.x;
    if (i >= n * HDIM) return;
    int f = i & (HDIM - 1);
    float inv_n = 1.0f / (float)n;
    float mu  = stats[f] * inv_n;
    float var = stats[HDIM + f] * inv_n - mu * mu;
    h[i] = (x[i] - mu) * rsqrtf(var + EPSV) * g[f] + b[f];
}

// ---------------- WMMA GEMM: m = act(in) @ W ; out = b + m*dis^2 ---------
// Block: 256 thr = 8 waves. Each wave -> 16 rows x 128 cols.
// V_WMMA_F32_16X16X4_F32, full f32 precision. mout may alias `in`
// (each wave reads then writes only its own 16 rows).
// W staged in LDS K-pair interleaved: Wp[j*WP_STRIDE + c*2 + t] = W[2j+t][c],
// so each lane's B fragment is one aligned ds_load_b64.
__global__ void k_gemm(const float* __restrict__ in, const float* __restrict__ W,
                       const float* __restrict__ bias, const float* __restrict__ dis,
                       float* __restrict__ mout, float* __restrict__ outp,
                       int nrows, int relu) {
    __shared__ float Wp[WP_STRIDE * (HDIM / 2)];          // 72 KB of 320 KB WGP LDS
    {
        const float4* src = reinterpret_cast<const float4*>(W);
        for (int i = threadIdx.x; i < HDIM * HDIM / 4; i += blockDim.x) {
            float4 v = src[i];
            int r = (i * 4) >> 7;                         // K row
            int c = (i * 4) & (HDIM - 1);                 // N col
            float* dst = &Wp[(r >> 1) * WP_STRIDE + c * 2 + (r & 1)];
            dst[0] = v.x; dst[2] = v.y; dst[4] = v.z; dst[6] = v.w;
        }
    }
    __syncthreads();

    const int wid  = threadIdx.x >> 5;
    const int lane = threadIdx.x & 31;
    const int half = lane >> 4;         // 0: lanes 0-15, 1: lanes 16-31
    const int l16  = lane & 15;
    const int row_base = blockIdx.x * 128 + wid * 16;
    const int row_a    = row_base + l16;                  // A-matrix: M = lane%16
    const bool aok     = row_a < nrows;
    // clamped address: invalid lanes read row 0 then select 0 (no exec dance)
    const float* ap = in + (size_t)(aok ? row_a : 0) * HDIM + half * 2;
    const float lim = relu ? 0.0f : -INFINITY;            // ReLU as one v_max

    v8f acc[8];
    const v8f zero = {0.f,0.f,0.f,0.f,0.f,0.f,0.f,0.f};
#pragma unroll
    for (int t = 0; t < 8; ++t) acc[t] = zero;

    for (int k = 0; k < HDIM; k += 4) {
        float a0 = fmaxf(ap[k],     lim);
        float a1 = fmaxf(ap[k + 1], lim);
        v2f a;
        a.x = aok ? a0 : 0.0f;
        a.y = aok ? a1 : 0.0f;

        const float* bp = &Wp[((k >> 1) + half) * WP_STRIDE + l16 * 2];
        v2f bfrag[8];
#pragma unroll
        for (int nt = 0; nt < 8; ++nt)
            bfrag[nt] = *reinterpret_cast<const v2f*>(bp + nt * 32);
#pragma unroll
        for (int nt = 0; nt < 8; ++nt)
            acc[nt] = __builtin_amdgcn_wmma_f32_16x16x4_f32(
                false, a, false, bfrag[nt], (short)0, acc[nt], false, false);
    }

    // C/D layout: VGPR r, lanes 0-15 -> M=r, lanes 16-31 -> M=r+8; N = lane%16
#pragma unroll
    for (int r = 0; r < 8; ++r) {
        int row = row_base + half * 8 + r;
        if (row < nrows) {
            float dd = dis[row]; dd *= dd;                // self-loop weight
            size_t rb = (size_t)row * HDIM;
#pragma unroll
            for (int nt = 0; nt < 8; ++nt) {
                int c = nt * 16 + l16;
                float v = acc[nt][r];
                mout[rb + c] = v;
                outp[rb + c] = bias[c] + v * dd;
            }
        }
    }
}

// ---------------- edge scatter: out[row] += m[col] * dis[row]*dis[col] ---
__global__ void k_edge(const float* __restrict__ m, const int* __restrict__ row,
                       const int* __restrict__ col, const float* __restrict__ dis,
                       float* __restrict__ outp, int E) {
    int w = (blockIdx.x * blockDim.x + threadIdx.x) >> 5;  // one wave per edge
    if (w >= E) return;
    int lane = threadIdx.x & 31;
    int r = row[w], c = col[w];
    float nw = dis[r] * dis[c];
    float4 v = reinterpret_cast<const float4*>(m + (size_t)c * HDIM)[lane];
    float* op = outp + (size_t)r * HDIM + lane * 4;
    atomicAdd(op + 0, v.x * nw);
    atomicAdd(op + 1, v.y * nw);
    atomicAdd(op + 2, v.z * nw);
    atomicAdd(op + 3, v.w * nw);
}

// ---------------- node attention + pooling (wave32 per node) -------------
__global__ void k_att(const float* __restrict__ h, const int* __restrict__ batch,
                      const float* __restrict__ Wna, const float* __restrict__ bna,
                      float* __restrict__ xc, float* __restrict__ xo,
                      float* __restrict__ cnt, int n) {
    int i = (blockIdx.x * blockDim.x + threadIdx.x) >> 5;
    if (i >= n) return;
    int lane = threadIdx.x & 31;
    int f0 = lane * 4;
    float4 hv = reinterpret_cast<const float4*>(h + (size_t)i * HDIM)[lane];
    hv.x = fmaxf(hv.x, 0.f); hv.y = fmaxf(hv.y, 0.f);     // final-layer ReLU
    hv.z = fmaxf(hv.z, 0.f); hv.w = fmaxf(hv.w, 0.f);

    float p0 = hv.x * Wna[(f0+0)*2] + hv.y * Wna[(f0+1)*2]
             + hv.z * Wna[(f0+2)*2] + hv.w * Wna[(f0+3)*2];
    float p1 = hv.x * Wna[(f0+0)*2+1] + hv.y * Wna[(f0+1)*2+1]
             + hv.z * Wna[(f0+2)*2+1] + hv.w * Wna[(f0+3)*2+1];
#pragma unroll
    for (int mk = 16; mk; mk >>= 1) {
        p0 += __shfl_xor(p0, mk, 32);
        p1 += __shfl_xor(p1, mk, 32);
    }
    p0 += bna[0]; p1 += bna[1];
    float mx = fmaxf(p0, p1);
    float e0 = __expf(p0 - mx), e1 = __expf(p1 - mx);
    float inv = 1.0f / (e0 + e1);
    float a0 = e0 * inv, a1 = e1 * inv;

    int g = batch[i];
    float* xcp = xc + (size_t)g * HDIM + f0;
    float* xop = xo + (size_t)g * HDIM + f0;
    atomicAdd(xcp + 0, a0 * hv.x); atomicAdd(xcp + 1, a0 * hv.y);
    atomicAdd(xcp + 2, a0 * hv.z); atomicAdd(xcp + 3, a0 * hv.w);
    atomicAdd(xop + 0, a1 * hv.x); atomicAdd(xop + 1, a1 * hv.y);
    atomicAdd(xop + 2, a1 * hv.z); atomicAdd(xop + 3, a1 * hv.w);
    if (lane == 0) atomicAdd(&cnt[g], 1.0f);
}

// ---------------- head: mean-pool norm -> BN(over G) -> Linear -----------
__global__ void k_head(const float* __restrict__ pool, const float* __restrict__ cnt,
                       const float* __restrict__ gamma, const float* __restrict__ beta,
                       const float* __restrict__ Wc, const float* __restrict__ bc,
                       float* __restrict__ out, int C) {
    __shared__ float s[GRP * HDIM];                       // 64 KB
    int f = threadIdx.x;                                  // 128 threads
    float sum = 0.f, sq = 0.f;
    for (int g = 0; g < GRP; ++g) {
        float v = pool[g * HDIM + f] / fmaxf(cnt[g], 1.0f);
        sum += v; sq += v * v;
    }
    float mu  = sum * (1.0f / (float)GRP);
    float var = sq  * (1.0f / (float)GRP) - mu * mu;
    float sc  = rsqrtf(var + EPSV) * gamma[f];
    float sh  = beta[f] - mu * sc;
    for (int g = 0; g < GRP; ++g) {
        float v = pool[g * HDIM + f] / fmaxf(cnt[g], 1.0f);
        s[g * HDIM + f] = v * sc + sh;
    }
    __syncthreads();
    int g = threadIdx.x;
    for (int c = 0; c < C; ++c) {
        float acc = bc[c];
        for (int ff = 0; ff < HDIM; ++ff) acc += s[g * HDIM + ff] * Wc[ff * C + c];
        out[g * C + c] = acc;
    }
}

// -------------------------------------------------------------------------
extern "C" void kernel_launch(void* const* d_in, const int* in_sizes, int n_in,
                              void* d_out, int out_size, void* d_ws, size_t ws_size,
                              hipStream_t stream) {
    const float* x        = (const float*)d_in[0];
    const int*   ei       = (const int*)d_in[1];
    const int*   batch    = (const int*)d_in[2];
    const float* bn_f_g   = (const float*)d_in[3];
    const float* bn_f_b   = (const float*)d_in[4];
    const float* W_feat   = (const float*)d_in[5];
    const float* b_feat   = (const float*)d_in[6];
    const float* W_convs  = (const float*)d_in[7];
    const float* b_convs  = (const float*)d_in[8];
    // d_in[9], d_in[10]: W_ea / b_ea -> edge attention is dead code, skipped
    const float* W_na     = (const float*)d_in[11];
    const float* b_na     = (const float*)d_in[12];
    const float* bn_c_g   = (const float*)d_in[13];
    const float* bn_c_b   = (const float*)d_in[14];
    const float* W_c      = (const float*)d_in[15];
    const float* b_c      = (const float*)d_in[16];
    const float* bn_o_g   = (const float*)d_in[17];
    const float* bn_o_b   = (const float*)d_in[18];
    const float* W_o      = (const float*)d_in[19];
    const float* b_o      = (const float*)d_in[20];
    float* out = (float*)d_out;

    const int N = in_sizes[0] / HDIM;
    const int E = in_sizes[1] / 2;
    const int C = in_sizes[16];          // b_c length = number of classes

    // workspace layout (floats), ~52 MB total
    float* ws    = (float*)d_ws;
    float* dis   = ws;                         // N   (degree then rsqrt)
    float* stats = dis + N;                    // 256
    float* cnt   = stats + 256;                // 128
    float* xc    = cnt + GRP;                  // 128*128
    float* xo    = xc + GRP * HDIM;            // 128*128
    float* hA    = xo + GRP * HDIM;            // N*128
    float* hB    = hA + (size_t)N * HDIM;      // N*128

    const int T = 256;
    k_init  <<<(N + T - 1) / T, T, 0, stream>>>(dis, stats, cnt, xc, xo, N, GRP * HDIM);
    k_degree<<<(E + T - 1) / T, T, 0, stream>>>(ei, dis, E);
    k_rsqrt <<<(N + T - 1) / T, T, 0, stream>>>(dis, N);
    k_bnstats<<<512, HDIM, 0, stream>>>(x, stats, N);
    k_bnapply<<<((size_t)N * HDIM + T - 1) / T, T, 0, stream>>>(x, stats, bn_f_g, bn_f_b, hA, N);

    const int gblocks = (N + 127) / 128;
    const int eblocks = (E + 7) / 8;

    // layer 0: in = bn(x) (no relu), m in-place in hA, out = hB
    k_gemm<<<gblocks, T, 0, stream>>>(hA, W_feat, b_feat, dis, hA, hB, N, 0);
    k_edge<<<eblocks, T, 0, stream>>>(hA, ei, ei + E, dis, hB, E);

    float* cur = hB;
    float* oth = hA;
    for (int l = 0; l < 3; ++l) {
        const float* Wl = W_convs + (size_t)l * HDIM * HDIM;
        const float* bl = b_convs + (size_t)l * HDIM;
        k_gemm<<<gblocks, T, 0, stream>>>(cur, Wl, bl, dis, cur, oth, N, 1);
        k_edge<<<eblocks, T, 0, stream>>>(cur, ei, ei + E, dis, oth, E);
        float* t = cur; cur = oth; oth = t;
    }
    // cur = final pre-activation h (ReLU fused into k_att)

    k_att<<<(N + 7) / 8, T, 0, stream>>>(cur, batch, W_na, b_na, xc, xo, cnt, N);
    k_head<<<1, HDIM, 0, stream>>>(xc, cnt, bn_c_g, bn_c_b, W_c, b_c, out, C);
    k_head<<<1, HDIM, 0, stream>>>(xo, cnt, bn_o_g, bn_o_b, W_o, b_o, out + GRP * C, C);
}